// XConvBatch_39685497815963
// MI455X (gfx1250) — compile-verified
//
#include <hip/hip_runtime.h>
#include <hip/hip_bf16.h>

// ---------------------------------------------------------------------------
// XConv (PointCNN-style) for MI455X / gfx1250, wave32.
// M=8192 points, K=16 neighbors, CIN=128, CD=32, C=160, DM=2, COUT=256.
// f32 WMMA (v_wmma_f32_16x16x4_f32) used for the per-point batched matmul
// x_star(160x16) @ t(16x16) and the final [8192,320]@[320,256] GEMM.
// ---------------------------------------------------------------------------

typedef __attribute__((ext_vector_type(2))) float v2f;
typedef __attribute__((ext_vector_type(8))) float v8f;

constexpr int MPTS  = 8192;      // N*I
constexpr int KNN   = 16;
constexpr int CINc  = 128;
constexpr int CDc   = 32;
constexpr int CCc   = 160;       // CIN + CD
constexpr int COUTc = 256;
constexpr int ROWS1 = MPTS * KNN; // 131072

__device__ __forceinline__ float elu1(float x) {
  return x > 0.f ? x : (expf(x) - 1.f);
}

// ---------------------------------------------------------------------------
// 1) KNN: one thread per query, LDS-tiled candidates, register top-16 (sorted
//    ascending; strict < keeps stable tie order, self (d=0) lands first).
// ---------------------------------------------------------------------------
__global__ __launch_bounds__(256) void knn_kernel(const float* __restrict__ pos,
                                                  int* __restrict__ idx) {
  const int q = blockIdx.x * 256 + threadIdx.x;
  const float qx = pos[q * 3 + 0], qy = pos[q * 3 + 1], qz = pos[q * 3 + 2];
  float bd[KNN];
  int   bi[KNN];
#pragma unroll
  for (int s = 0; s < KNN; ++s) { bd[s] = 3.4e38f; bi[s] = 0; }

  __shared__ float tp[256 * 3];
  for (int t = 0; t < MPTS; t += 256) {
    __syncthreads();
    for (int i = threadIdx.x; i < 256 * 3; i += 256) tp[i] = pos[t * 3 + i];
    __syncthreads();
    for (int c = 0; c < 256; ++c) {
      const float dx = tp[c * 3 + 0] - qx;
      const float dy = tp[c * 3 + 1] - qy;
      const float dz = tp[c * 3 + 2] - qz;
      const float d  = dx * dx + dy * dy + dz * dz;
      if (d < bd[KNN - 1]) {
        float cd = d; int ci = t + c;
#pragma unroll
        for (int s = 0; s < KNN; ++s) {
          const bool sw = cd < bd[s];
          const float td = sw ? bd[s] : cd;
          const int   ti = sw ? bi[s] : ci;
          bd[s] = sw ? cd : bd[s];
          bi[s] = sw ? ci : bi[s];
          cd = td; ci = ti;
        }
      }
    }
  }
#pragma unroll
  for (int s = 0; s < KNN; ++s) idx[q * KNN + s] = bi[s];
}

// ---------------------------------------------------------------------------
// 2) rel + mlp1 layer A: per (m,k) row -> rel[3], a1 = elu(rel@w1a + b1a)
// ---------------------------------------------------------------------------
__global__ __launch_bounds__(256) void rel_mlp1a_kernel(
    const float* __restrict__ pos, const int* __restrict__ idx,
    const float* __restrict__ w1a, const float* __restrict__ b1a,
    float* __restrict__ rel, float* __restrict__ a1) {
  const int row = blockIdx.x * 256 + threadIdx.x;   // 0..131071
  const int m = row >> 4;
  const int j = idx[row];
  const float rx = pos[j * 3 + 0] - pos[m * 3 + 0];
  const float ry = pos[j * 3 + 1] - pos[m * 3 + 1];
  const float rz = pos[j * 3 + 2] - pos[m * 3 + 2];
  rel[row * 3 + 0] = rx; rel[row * 3 + 1] = ry; rel[row * 3 + 2] = rz;
#pragma unroll
  for (int c = 0; c < CDc; ++c) {
    const float z = rx * w1a[c] + ry * w1a[32 + c] + rz * w1a[64 + c] + b1a[c];
    a1[row * CDc + c] = elu1(z);
  }
}

// ---------------------------------------------------------------------------
// BN helpers: column sum/sumsq reduction (1 block/channel) + finalize to
// per-channel (scale, shift) with biased variance, eps = 1e-5.
// ---------------------------------------------------------------------------
__global__ __launch_bounds__(256) void bn_stats_kernel(
    const float* __restrict__ x, int rows, int cols, float* __restrict__ sums) {
  const int c = blockIdx.x;
  float s = 0.f, s2 = 0.f;
  for (int r = threadIdx.x; r < rows; r += 256) {
    const float v = x[(size_t)r * cols + c];
    s += v; s2 += v * v;
  }
  __shared__ float ls[256], ls2[256];
  ls[threadIdx.x] = s; ls2[threadIdx.x] = s2;
  __syncthreads();
  for (int o = 128; o > 0; o >>= 1) {
    if (threadIdx.x < o) {
      ls[threadIdx.x]  += ls[threadIdx.x + o];
      ls2[threadIdx.x] += ls2[threadIdx.x + o];
    }
    __syncthreads();
  }
  if (threadIdx.x == 0) { sums[c] = ls[0]; sums[cols + c] = ls2[0]; }
}

__global__ void bn_finalize_kernel(const float* __restrict__ sums,
                                   const float* __restrict__ g,
                                   const float* __restrict__ be,
                                   int rows, int cols, float* __restrict__ ss) {
  const int c = blockIdx.x * blockDim.x + threadIdx.x;
  if (c >= cols) return;
  const float inv  = 1.f / (float)rows;
  const float mean = sums[c] * inv;
  const float var  = sums[cols + c] * inv - mean * mean;
  const float sc   = g[c] * rsqrtf(var + 1e-5f);
  ss[c]        = sc;
  ss[cols + c] = be[c] - mean * sc;
}

// ---------------------------------------------------------------------------
// 3) mlp1 layer B: a2 = elu(bn1a(a1) @ w1b + b1b)
// ---------------------------------------------------------------------------
__global__ __launch_bounds__(256) void mlp1b_kernel(
    const float* __restrict__ a1, const float* __restrict__ ss1a,
    const float* __restrict__ w1b, const float* __restrict__ b1b,
    float* __restrict__ a2) {
  const int row = blockIdx.x * 256 + threadIdx.x;
  float h[CDc];
#pragma unroll
  for (int c = 0; c < CDc; ++c)
    h[c] = a1[row * CDc + c] * ss1a[c] + ss1a[CDc + c];
#pragma unroll 4
  for (int c2 = 0; c2 < CDc; ++c2) {
    float z = b1b[c2];
#pragma unroll
    for (int c = 0; c < CDc; ++c) z += h[c] * w1b[c * CDc + c2];
    a2[row * CDc + c2] = elu1(z);
  }
}

// ---------------------------------------------------------------------------
// 4) mlp2 stage 1: q1 = elu(rel[m][48] @ w2[48,256] + b2)  (1 block / point)
// ---------------------------------------------------------------------------
__global__ __launch_bounds__(256) void t1_kernel(
    const float* __restrict__ rel, const float* __restrict__ w2,
    const float* __restrict__ b2, float* __restrict__ q1) {
  const int m = blockIdx.x;
  __shared__ float r48[48];
  if (threadIdx.x < 48) r48[threadIdx.x] = rel[m * 48 + threadIdx.x];
  __syncthreads();
  const int c = threadIdx.x;
  float z = b2[c];
#pragma unroll 4
  for (int i = 0; i < 48; ++i) z += r48[i] * w2[i * 256 + c];
  q1[m * 256 + c] = elu1(z);
}

// ---------------------------------------------------------------------------
// 5) grouped conv: out[c= cc*16+o] = act( sum_l bn(qin)[cc*16+l]*wc[cc][o][l] + bc[c] )
// ---------------------------------------------------------------------------
__global__ __launch_bounds__(256) void gconv_kernel(
    const float* __restrict__ qin, const float* __restrict__ ss,
    const float* __restrict__ wc, const float* __restrict__ bc,
    float* __restrict__ qout, int do_elu) {
  const int m = blockIdx.x;
  __shared__ float tt[256];
  const int c = threadIdx.x;
  tt[c] = qin[m * 256 + c] * ss[c] + ss[256 + c];
  __syncthreads();
  const int cc = c >> 4, o = c & 15;
  float z = bc[c];
#pragma unroll
  for (int l = 0; l < 16; ++l) z += tt[cc * 16 + l] * wc[cc * 256 + o * 16 + l];
  qout[m * 256 + c] = do_elu ? elu1(z) : z;
}

// ---------------------------------------------------------------------------
// 6) xconv core: build x_star[160][16] + t[16][16] in LDS, per-wave 16x16
//    tile of xt via 4x v_wmma_f32_16x16x4_f32, then depthwise wcd collapse -> y.
//    blockDim = 320 (10 waves), one block per point.
// ---------------------------------------------------------------------------
__global__ __launch_bounds__(320) void xconv_core_kernel(
    const float* __restrict__ xf, const float* __restrict__ a2,
    const float* __restrict__ ss1b, const float* __restrict__ q3,
    const float* __restrict__ ss2c, const int* __restrict__ idx,
    const float* __restrict__ wcd, const float* __restrict__ bcd,
    float* __restrict__ y) {
  const int m = blockIdx.x;
  const int tid = threadIdx.x;
  __shared__ float xs[CCc * 16];   // x_star[c][k]
  __shared__ float tts[256];       // t[k][l]
  __shared__ float xts[CCc * 16];  // xt[c][l]
  __shared__ int ji[16];

  if (tid < 16) ji[tid] = idx[m * 16 + tid];
  __syncthreads();

  // t_final = bn2c(q3[m])
  for (int i = tid; i < 256; i += 320)
    tts[i] = q3[m * 256 + i] * ss2c[i] + ss2c[256 + i];
  // channels 0..31: h = bn1b(a2), xs[c*16+k]
  for (int i = tid; i < 512; i += 320) {
    const int c = i >> 4, k = i & 15;
    xs[i] = a2[(m * 16 + k) * CDc + c] * ss1b[c] + ss1b[CDc + c];
  }
  // channels 32..159: gathered xf rows (coalesced along feature dim)
  for (int i = tid; i < 2048; i += 320) {
    const int k = i >> 7, cc = i & 127;
    xs[(32 + cc) * 16 + k] = xf[ji[k] * CINc + cc];
  }
  __syncthreads();

  // per-wave 16x16 tile of xt = xs @ t, K-dim 16 -> 4 WMMA steps
  const int wv = tid >> 5, lane = tid & 31;
  const int r = lane & 15;
  const int half = (lane < 16) ? 0 : 2;
  v8f acc = {};
#pragma unroll
  for (int kk = 0; kk < 16; kk += 4) {
    v2f a, b;
    a.x = xs[(wv * 16 + r) * 16 + kk + half];
    a.y = xs[(wv * 16 + r) * 16 + kk + half + 1];
    b.x = tts[(kk + half) * 16 + r];
    b.y = tts[(kk + half + 1) * 16 + r];
    acc = __builtin_amdgcn_wmma_f32_16x16x4_f32(false, a, false, b,
                                                (short)0, acc, false, false);
  }
#pragma unroll
  for (int r8 = 0; r8 < 8; ++r8) {
    const int rowc = wv * 16 + r8 + ((lane < 16) ? 0 : 8);
    xts[rowc * 16 + r] = acc[r8];
  }
  __syncthreads();

  // y[m, 2c+o] = sum_l xt[c][l] * wcd[c][o][l] + bcd[2c+o]
  if (tid < CCc) {
    const int c = tid;
    float y0 = bcd[2 * c], y1 = bcd[2 * c + 1];
#pragma unroll
    for (int l = 0; l < 16; ++l) {
      const float v = xts[c * 16 + l];
      y0 += v * wcd[c * 32 + l];
      y1 += v * wcd[c * 32 + 16 + l];
    }
    y[m * 320 + 2 * c]     = y0;
    y[m * 320 + 2 * c + 1] = y1;
  }
}

// ---------------------------------------------------------------------------
// 7) final GEMM: out[8192,256] = Y[8192,320] @ wl[320,256] + bl
//    one 16x16 output tile per wave, 80 f32 WMMAs over k.
// ---------------------------------------------------------------------------
__global__ __launch_bounds__(256) void final_gemm_kernel(
    const float* __restrict__ Y, const float* __restrict__ wl,
    const float* __restrict__ bl, float* __restrict__ out) {
  const int wv = threadIdx.x >> 5, lane = threadIdx.x & 31;
  const int tile = blockIdx.x * 8 + wv;     // 8192 tiles total
  const int m0 = (tile >> 4) * 16;
  const int n0 = (tile & 15) * 16;
  const int r = lane & 15;
  const int half = (lane < 16) ? 0 : 2;
  v8f acc = {};
  for (int kk = 0; kk < 320; kk += 4) {
    v2f a, b;
    const float* ap = Y + (m0 + r) * 320 + kk + half;
    a.x = ap[0];
    a.y = ap[1];
    b.x = wl[(kk + half) * 256 + n0 + r];
    b.y = wl[(kk + half + 1) * 256 + n0 + r];
    acc = __builtin_amdgcn_wmma_f32_16x16x4_f32(false, a, false, b,
                                                (short)0, acc, false, false);
  }
  const float bias = bl[n0 + r];
#pragma unroll
  for (int r8 = 0; r8 < 8; ++r8) {
    const int row = m0 + r8 + ((lane < 16) ? 0 : 8);
    out[row * 256 + n0 + r] = acc[r8] + bias;
  }
}

// ---------------------------------------------------------------------------
// Host launch
// ---------------------------------------------------------------------------
extern "C" void kernel_launch(void* const* d_in, const int* in_sizes, int n_in,
                              void* d_out, int out_size, void* d_ws, size_t ws_size,
                              hipStream_t stream) {
  (void)in_sizes; (void)n_in; (void)out_size; (void)ws_size;
  const float* x    = (const float*)d_in[0];   // [8192,128]
  const float* pos  = (const float*)d_in[1];   // [8192,3]
  const float* w1a  = (const float*)d_in[2];
  const float* b1a  = (const float*)d_in[3];
  const float* g1a  = (const float*)d_in[4];
  const float* be1a = (const float*)d_in[5];
  const float* w1b  = (const float*)d_in[6];
  const float* b1b  = (const float*)d_in[7];
  const float* g1b  = (const float*)d_in[8];
  const float* be1b = (const float*)d_in[9];
  const float* w2   = (const float*)d_in[10];
  const float* b2   = (const float*)d_in[11];
  const float* g2a  = (const float*)d_in[12];
  const float* be2a = (const float*)d_in[13];
  const float* wc2a = (const float*)d_in[14];
  const float* bc2a = (const float*)d_in[15];
  const float* g2b  = (const float*)d_in[16];
  const float* be2b = (const float*)d_in[17];
  const float* wc2b = (const float*)d_in[18];
  const float* bc2b = (const float*)d_in[19];
  const float* g2c  = (const float*)d_in[20];
  const float* be2c = (const float*)d_in[21];
  const float* wcd  = (const float*)d_in[22];
  const float* bcd  = (const float*)d_in[23];
  const float* wl   = (const float*)d_in[24];
  const float* bl   = (const float*)d_in[25];
  float* out = (float*)d_out;

  // workspace carve-up (~72 MB total; fits in the 192 MB L2)
  char* w = (char*)d_ws;
  size_t off = 0;
  auto carve = [&](size_t bytes) { char* p = w + off; off += (bytes + 255) & ~(size_t)255; return p; };
  int*   idxb = (int*)  carve((size_t)ROWS1 * 4);            // 512 KB
  float* rel  = (float*)carve((size_t)ROWS1 * 3 * 4);        // 1.5 MB
  float* a1   = (float*)carve((size_t)ROWS1 * CDc * 4);      // 16 MB
  float* a2   = (float*)carve((size_t)ROWS1 * CDc * 4);      // 16 MB
  float* q1   = (float*)carve((size_t)MPTS * 256 * 4);       // 8 MB
  float* q2   = (float*)carve((size_t)MPTS * 256 * 4);       // 8 MB
  float* q3   = (float*)carve((size_t)MPTS * 256 * 4);       // 8 MB
  float* yb   = (float*)carve((size_t)MPTS * 320 * 4);       // 10 MB
  float* sums = (float*)carve(512 * 4);
  float* ss1a = (float*)carve(64 * 4);
  float* ss1b = (float*)carve(64 * 4);
  float* ss2a = (float*)carve(512 * 4);
  float* ss2b = (float*)carve(512 * 4);
  float* ss2c = (float*)carve(512 * 4);

  // 1) KNN
  knn_kernel<<<MPTS / 256, 256, 0, stream>>>(pos, idxb);
  // 2) rel + mlp1 layer A, BN stats
  rel_mlp1a_kernel<<<ROWS1 / 256, 256, 0, stream>>>(pos, idxb, w1a, b1a, rel, a1);
  bn_stats_kernel<<<CDc, 256, 0, stream>>>(a1, ROWS1, CDc, sums);
  bn_finalize_kernel<<<1, CDc, 0, stream>>>(sums, g1a, be1a, ROWS1, CDc, ss1a);
  // 3) mlp1 layer B, BN stats (applied later in xconv_core)
  mlp1b_kernel<<<ROWS1 / 256, 256, 0, stream>>>(a1, ss1a, w1b, b1b, a2);
  bn_stats_kernel<<<CDc, 256, 0, stream>>>(a2, ROWS1, CDc, sums);
  bn_finalize_kernel<<<1, CDc, 0, stream>>>(sums, g1b, be1b, ROWS1, CDc, ss1b);
  // 4) mlp2 stage 1
  t1_kernel<<<MPTS, 256, 0, stream>>>(rel, w2, b2, q1);
  bn_stats_kernel<<<256, 256, 0, stream>>>(q1, MPTS, 256, sums);
  bn_finalize_kernel<<<1, 256, 0, stream>>>(sums, g2a, be2a, MPTS, 256, ss2a);
  // 5) grouped convs
  gconv_kernel<<<MPTS, 256, 0, stream>>>(q1, ss2a, wc2a, bc2a, q2, 1);
  bn_stats_kernel<<<256, 256, 0, stream>>>(q2, MPTS, 256, sums);
  bn_finalize_kernel<<<1, 256, 0, stream>>>(sums, g2b, be2b, MPTS, 256, ss2b);
  gconv_kernel<<<MPTS, 256, 0, stream>>>(q2, ss2b, wc2b, bc2b, q3, 0);
  bn_stats_kernel<<<256, 256, 0, stream>>>(q3, MPTS, 256, sums);
  bn_finalize_kernel<<<1, 256, 0, stream>>>(sums, g2c, be2c, MPTS, 256, ss2c);
  // 6) x_star @ t (WMMA) + depthwise collapse -> y
  xconv_core_kernel<<<MPTS, 320, 0, stream>>>(x, a2, ss1b, q3, ss2c, idxb,
                                              wcd, bcd, yb);
  // 7) final GEMM (WMMA) + bias
  final_gemm_kernel<<<MPTS * (COUTc / 16) / (16 * 8), 256, 0, stream>>>(yb, wl, bl, out);
}